// Huber_loss_20255065768293
// MI455X (gfx1250) — compile-verified
//
#include <hip/hip_runtime.h>

#define TAU 0.1f

typedef __attribute__((ext_vector_type(4))) float f4;
typedef __attribute__((ext_vector_type(2))) float v2f;
typedef __attribute__((ext_vector_type(8))) float v8f;

__device__ __forceinline__ float huber(float e) {
    float a    = __builtin_fabsf(e);
    float quad = 0.5f * e * e;
    // tau*(a-tau) + tau^2/2  ==  tau*a - tau^2/2
    float lin  = __builtin_fmaf(TAU, a, -0.5f * TAU * TAU);
    return (a <= TAU) ? quad : lin;
}

// Sum 64 f32 values (2 per lane, 32 lanes) with one V_WMMA_F32_16X16X4_F32.
// A = all-ones 16x4, B = {acc0, acc1} per lane. D[m][n] = colsum[n]; per the
// 16x16 f32 C/D layout, lane L's c[0] holds colsum[L%16] (columns duplicated
// in lanes n and n+16), so xor-reduction over masks 1,2,4,8 yields the total
// (= sum of ALL B entries, independent of the exact B lane mapping) in every
// lane. EXEC must be all ones at the call site.
__device__ __forceinline__ float wave_sum_wmma(float acc0, float acc1) {
    v2f a; a.x = 1.0f; a.y = 1.0f;
    v2f b; b.x = acc0; b.y = acc1;
    v8f c = {};
    c = __builtin_amdgcn_wmma_f32_16x16x4_f32(
        /*neg_a=*/false, a, /*neg_b=*/false, b,
        /*c_mod=*/(short)0, c, /*reuse_a=*/false, /*reuse_b=*/false);
    float s = c[0];
    s += __shfl_xor(s, 1, 32);
    s += __shfl_xor(s, 2, 32);
    s += __shfl_xor(s, 4, 32);
    s += __shfl_xor(s, 8, 32);
    return s;
}

// Pass 1: streaming Huber + per-block partial sums. Bandwidth-bound:
// non-temporal b128 loads, grid-stride, 1M threads in flight.
__global__ void __launch_bounds__(256)
huber_partial_kernel(const f4* __restrict__ in, float* __restrict__ partial,
                     int n4, int n) {
    const int tid    = blockIdx.x * blockDim.x + threadIdx.x;
    const int stride = gridDim.x * blockDim.x;

    float acc0 = 0.0f, acc1 = 0.0f;
    for (int i = tid; i < n4; i += stride) {
        f4 v = __builtin_nontemporal_load(&in[i]);
        acc0 += huber(v.x) + huber(v.y);
        acc1 += huber(v.z) + huber(v.w);
    }
    // Scalar tail (n not multiple of 4) — reconverges before the WMMA below.
    if (tid == 0) {
        const float* sf = (const float*)in;
        for (int i = n4 * 4; i < n; ++i) acc0 += huber(sf[i]);
    }

    float wsum = wave_sum_wmma(acc0, acc1);

    __shared__ float lds[8];
    const int lane = threadIdx.x & 31;
    const int wave = threadIdx.x >> 5;
    if (lane == 0) lds[wave] = wsum;
    __syncthreads();
    if (threadIdx.x == 0) {
        float bsum = 0.0f;
        const int nwaves = blockDim.x >> 5;
        for (int w = 0; w < nwaves; ++w) bsum += lds[w];
        partial[blockIdx.x] = bsum;
    }
}

// Pass 2: one wave reduces the block partials, chaining WMMAs through the
// C operand (64 values folded per matrix op, exact f32 accumulation).
__global__ void __launch_bounds__(32)
huber_final_kernel(const float* __restrict__ partial, float* __restrict__ out,
                   int nb) {
    const int lane = threadIdx.x;
    v2f a; a.x = 1.0f; a.y = 1.0f;
    v8f c = {};
    for (int i = 0; i < nb; i += 64) {
        const int i0 = i + lane;
        const int i1 = i + 32 + lane;
        v2f b;
        b.x = (i0 < nb) ? partial[i0] : 0.0f;
        b.y = (i1 < nb) ? partial[i1] : 0.0f;
        c = __builtin_amdgcn_wmma_f32_16x16x4_f32(
            false, a, false, b, (short)0, c, false, false);
    }
    float s = c[0];
    s += __shfl_xor(s, 1, 32);
    s += __shfl_xor(s, 2, 32);
    s += __shfl_xor(s, 4, 32);
    s += __shfl_xor(s, 8, 32);
    if (lane == 0) out[0] = s;
}

extern "C" void kernel_launch(void* const* d_in, const int* in_sizes, int n_in,
                              void* d_out, int out_size, void* d_ws, size_t ws_size,
                              hipStream_t stream) {
    (void)n_in; (void)out_size;
    const f4*  in      = (const f4*)d_in[0];
    float*     partial = (float*)d_ws;
    float*     out     = (float*)d_out;
    const int  n  = in_sizes[0];
    const int  n4 = n >> 2;

    int blocks = 4096;                              // 16 KiB of partials
    if (ws_size < (size_t)blocks * sizeof(float))   // deterministic fallback
        blocks = (int)(ws_size / sizeof(float));
    if (blocks < 1) blocks = 1;

    huber_partial_kernel<<<blocks, 256, 0, stream>>>(in, partial, n4, n);
    huber_final_kernel<<<1, 32, 0, stream>>>(partial, out, blocks);
}